// LSTMForecast_9972914061635
// MI455X (gfx1250) — compile-verified
//
#include <hip/hip_runtime.h>
#include <hip/hip_bf16.h>
#include <math.h>

typedef __attribute__((ext_vector_type(2))) float v2f;
typedef __attribute__((ext_vector_type(8))) float v8f;

// Problem constants (match reference)
#define BSZ   256
#define TLEN  512
#define HDIM  256
#define GDIM  1024   // 4*H

// LDS padded strides (floats) chosen for bank-conflict-free WMMA fragment reads
#define HP 260       // h rows:  (m*260+k)%64 = (4m+k)%64 -> distinct across 16 lanes
#define GP 1028      // gates rows

__device__ __forceinline__ float sigmoidf_fast(float x) {
    return 1.0f / (1.0f + __expf(-x));
}

// One LSTM layer, persistent over the whole time scan.
// Grid: B/16 workgroups; Block: 512 threads (16 waves).
// Each workgroup owns batch rows [b0, b0+16); each wave owns 64 gate columns.
// gates = x_t @ W_ih^T + h @ W_hh^T + (b_ih + b_hh), computed with
// v_wmma_f32_16x16x4_f32 (f32 in/out, reference precision).
template <int IN_K, bool STORE_ALL>
__global__ __launch_bounds__(512) void lstm_layer_kernel(
    const float* __restrict__ x,     // [B, T, IN_K]
    const float* __restrict__ Wih,   // [4H, IN_K]
    const float* __restrict__ Whh,   // [4H, H]
    const float* __restrict__ bih,   // [4H]
    const float* __restrict__ bhh,   // [4H]
    float* __restrict__ hout)        // STORE_ALL ? [B,T,H] : [B,H]
{
    static_assert(IN_K % 4 == 0, "K must be multiple of WMMA K=4");
    constexpr int XP = (IN_K == 32) ? 36 : HP;   // padded x-tile stride

    extern __shared__ float smem[];
    float* s_bias = smem;                    // [GDIM]
    float* s_h    = s_bias + GDIM;           // [16][HP]
    float* s_c    = s_h + 16 * HP;           // [16][HDIM]
    float* s_x    = s_c + 16 * HDIM;         // [16][XP]
    float* s_g    = s_x + 16 * XP;           // [16][GP]

    const int tid  = threadIdx.x;
    const int lane = tid & 31;
    const int wave = tid >> 5;               // 0..15
    const int m16  = lane & 15;              // row (A) / col (B) within tile
    const int half = lane >> 4;              // 0/1 -> K sub-offset 2*half

    // --- init: combined bias, h=c=0 ---
    for (int i = tid; i < GDIM; i += 512)      s_bias[i] = bih[i] + bhh[i];
    for (int i = tid; i < 16 * HP; i += 512)   s_h[i] = 0.0f;
    for (int i = tid; i < 16 * HDIM; i += 512) s_c[i] = 0.0f;
    __syncthreads();

    const int b0    = blockIdx.x * 16;
    const int ncol0 = wave * 64;             // this wave's first gate column

    for (int t = 0; t < TLEN; ++t) {
        // --- stage x_t tile (16 x IN_K) into LDS ---
        for (int i = tid; i < 16 * IN_K; i += 512) {
            const int m = i / IN_K, kk = i % IN_K;
            s_x[m * XP + kk] = x[((size_t)(b0 + m) * TLEN + t) * IN_K + kk];
        }
        __syncthreads();

        v8f acc[4] = {};   // 4 N-tiles of 16 columns, f32 accumulators

        // --- input projection: A = x_t (LDS), B = W_ih^T (global) ---
        for (int k = 0; k < IN_K; k += 4) {
            const v2f a = *(const v2f*)&s_x[m16 * XP + k + 2 * half];
#pragma unroll
            for (int j = 0; j < 4; ++j) {
                const int n = ncol0 + j * 16 + m16;
                const v2f b = *(const v2f*)&Wih[(size_t)n * IN_K + k + 2 * half];
                acc[j] = __builtin_amdgcn_wmma_f32_16x16x4_f32(
                    false, a, false, b, (short)0, acc[j], false, false);
            }
        }

        // --- recurrent projection: A = h (LDS), B = W_hh^T (global, L2-resident) ---
        for (int k = 0; k < HDIM; k += 4) {
            const v2f a = *(const v2f*)&s_h[m16 * HP + k + 2 * half];
#pragma unroll
            for (int j = 0; j < 4; ++j) {
                const int n = ncol0 + j * 16 + m16;
                const v2f b = *(const v2f*)&Whh[(size_t)n * HDIM + k + 2 * half];
                acc[j] = __builtin_amdgcn_wmma_f32_16x16x4_f32(
                    false, a, false, b, (short)0, acc[j], false, false);
            }
        }

        // --- bias add + spill pre-activations to LDS gate buffer ---
#pragma unroll
        for (int j = 0; j < 4; ++j) {
            const int   n  = ncol0 + j * 16 + m16;
            const float bs = s_bias[n];
#pragma unroll
            for (int r = 0; r < 8; ++r) {
                const int mrow = r + 8 * half;   // D-matrix VGPR layout
                s_g[mrow * GP + n] = acc[j][r] + bs;
            }
        }
        __syncthreads();

        // --- elementwise LSTM cell update: 4096 elems, 8 per thread ---
#pragma unroll
        for (int e = 0; e < 8; ++e) {
            const int idx = tid * 8 + e;
            const int m   = idx >> 8;
            const int hh  = idx & 255;
            const float gi = s_g[m * GP + hh];
            const float gf = s_g[m * GP + 256 + hh];
            const float gg = s_g[m * GP + 512 + hh];
            const float go = s_g[m * GP + 768 + hh];
            const float i_ = sigmoidf_fast(gi);
            const float f_ = sigmoidf_fast(gf);
            const float g_ = tanhf(gg);
            const float o_ = sigmoidf_fast(go);
            const float c  = f_ * s_c[m * HDIM + hh] + i_ * g_;
            const float h  = o_ * tanhf(c);
            s_c[m * HDIM + hh] = c;
            s_h[m * HP + hh]   = h;
            if (STORE_ALL) {
                hout[((size_t)(b0 + m) * TLEN + t) * HDIM + hh] = h;
            } else if (t == TLEN - 1) {
                hout[(size_t)(b0 + m) * HDIM + hh] = h;
            }
        }
        __syncthreads();   // h/c visible before next step's WMMA reads
    }
}

// out[b] = dot(h_last[b,:], W_fc[0,:]) + b_fc
__global__ __launch_bounds__(256) void fc_kernel(
    const float* __restrict__ hlast,  // [B, H]
    const float* __restrict__ Wfc,    // [1, H]
    const float* __restrict__ bfc,    // [1]
    float* __restrict__ out)          // [B, 1]
{
    __shared__ float red[256];
    const int b   = blockIdx.x;
    const int tid = threadIdx.x;
    red[tid] = hlast[(size_t)b * HDIM + tid] * Wfc[tid];
    __syncthreads();
    for (int s = 128; s > 0; s >>= 1) {
        if (tid < s) red[tid] += red[tid + s];
        __syncthreads();
    }
    if (tid == 0) out[b] = red[0] + bfc[0];
}

extern "C" void kernel_launch(void* const* d_in, const int* in_sizes, int n_in,
                              void* d_out, int out_size, void* d_ws, size_t ws_size,
                              hipStream_t stream) {
    const float* x     = (const float*)d_in[0];
    const float* Wih0  = (const float*)d_in[1];
    const float* Whh0  = (const float*)d_in[2];
    const float* bih0  = (const float*)d_in[3];
    const float* bhh0  = (const float*)d_in[4];
    const float* Wih1  = (const float*)d_in[5];
    const float* Whh1  = (const float*)d_in[6];
    const float* bih1  = (const float*)d_in[7];
    const float* bhh1  = (const float*)d_in[8];
    const float* Wfc   = (const float*)d_in[9];
    const float* bfc   = (const float*)d_in[10];
    float* out = (float*)d_out;

    // Workspace: layer-0 hidden sequence + layer-1 last hidden
    float* h1s = (float*)d_ws;                          // [B, T, H] = 134 MB
    float* h2l = h1s + (size_t)BSZ * TLEN * HDIM;       // [B, H]

    constexpr size_t smem0 =
        (GDIM + 16 * HP + 16 * HDIM + 16 * 36 + 16 * GP) * sizeof(float); // ~105 KB
    constexpr size_t smem1 =
        (GDIM + 16 * HP + 16 * HDIM + 16 * HP + 16 * GP) * sizeof(float); // ~120 KB

    lstm_layer_kernel<32, true><<<BSZ / 16, 512, smem0, stream>>>(
        x, Wih0, Whh0, bih0, bhh0, h1s);
    lstm_layer_kernel<256, false><<<BSZ / 16, 512, smem1, stream>>>(
        h1s, Wih1, Whh1, bih1, bhh1, h2l);
    fc_kernel<<<BSZ, 256, 0, stream>>>(h2l, Wfc, bfc, out);
}